// ScalarHead_84361747628504
// MI455X (gfx1250) — compile-verified
//
#include <hip/hip_runtime.h>
#include <hip/hip_bf16.h>

typedef __attribute__((ext_vector_type(2))) float v2f;
typedef __attribute__((ext_vector_type(4))) float v4f;
typedef __attribute__((ext_vector_type(8))) float v8f;

#define D_MODEL 128
#define HIDDEN  64
#define LDS_STRIDE 68   // 16x64 tile padded to avoid bank conflicts

// first index i in [0,n) with ids[i] >= v  (ids sorted ascending)
__device__ __forceinline__ int lower_bound(const int* __restrict__ ids, int n, int v) {
    int lo = 0, hi = n;
    while (lo < hi) {
        int mid = (lo + hi) >> 1;
        if (ids[mid] < v) lo = mid + 1; else hi = mid;
    }
    return lo;
}

// One wave32 per segment: binary-search the contiguous run, stream rows,
// each lane accumulates 4 of the 128 dims as float4 (coalesced 512B/row).
__global__ void __launch_bounds__(256)
pool_kernel(const float* __restrict__ h, const int* __restrict__ ids,
            float* __restrict__ pooled, int n_total, int num_segments) {
    const int wave = (blockIdx.x * blockDim.x + threadIdx.x) >> 5;
    const int lane = threadIdx.x & 31;
    if (wave >= num_segments) return;

    const int start = lower_bound(ids, n_total, wave);
    const int end   = lower_bound(ids, n_total, wave + 1);

    v4f s0 = {0.f, 0.f, 0.f, 0.f}, s1 = s0, s2 = s0, s3 = s0;
    const float* base = h + (size_t)start * D_MODEL + lane * 4;
    int r = start;
    for (; r + 4 <= end; r += 4) {
        v4f x0 = *(const v4f*)(base + 0 * D_MODEL);
        v4f x1 = *(const v4f*)(base + 1 * D_MODEL);
        v4f x2 = *(const v4f*)(base + 2 * D_MODEL);
        v4f x3 = *(const v4f*)(base + 3 * D_MODEL);
        s0 += x0; s1 += x1; s2 += x2; s3 += x3;
        base += 4 * D_MODEL;
    }
    for (; r < end; ++r) {
        s0 += *(const v4f*)base;
        base += D_MODEL;
    }
    v4f sum = (s0 + s1) + (s2 + s3);
    const float inv = (end > start) ? 1.0f / (float)(end - start) : 0.0f;
    sum *= inv;
    *(v4f*)(pooled + (size_t)wave * D_MODEL + lane * 4) = sum;
}

__device__ __forceinline__ float silu(float x) {
    return x / (1.0f + __expf(-x));
}

// One wave32 per 16-molecule tile. Full-precision f32 WMMA (16x16x4).
// A frag (16x4): lane = (half<<4)|sub holds A[sub][kb], A[sub][kb+1], kb=2*half
// B frag (4x16): same lane holds B[kb][sub], B[kb+1][sub]
// C/D  (16x16): reg i at that lane is C[i + 8*half][sub]
__global__ void __launch_bounds__(256)
mlp_kernel(const float* __restrict__ pooled,
           const float* __restrict__ W1, const float* __restrict__ b1,
           const float* __restrict__ W2, const float* __restrict__ b2,
           const float* __restrict__ W3, const float* __restrict__ b3,
           float* __restrict__ out, int num_tiles, int B) {
    __shared__ float lds[8 * 16 * LDS_STRIDE];
    const int waveInBlock = threadIdx.x >> 5;
    const int lane        = threadIdx.x & 31;
    const int tile        = blockIdx.x * 8 + waveInBlock;
    if (tile >= num_tiles) return;            // whole-wave exit: EXEC stays all-1s

    float* X = lds + waveInBlock * 16 * LDS_STRIDE;   // wave-private [16][68]
    const int half = lane >> 4;               // selects K pair / upper rows
    const int sub  = lane & 15;               // A row / B,C column

    int arow = tile * 16 + sub;
    if (arow >= B) arow = B - 1;              // safe clamp (B % 16 == 0 normally)
    const float* prow = pooled + (size_t)arow * D_MODEL;

    v8f acc[4];

    // ---------------- Layer 1: [16x128] @ [128x64] + b1, SiLU ----------------
#pragma unroll
    for (int nt = 0; nt < 4; ++nt) {
        const float bv = b1[nt * 16 + sub];
#pragma unroll
        for (int i = 0; i < 8; ++i) acc[nt][i] = bv;
    }
#pragma unroll 4
    for (int ks = 0; ks < 32; ++ks) {
        const int kb = ks * 4 + half * 2;
        v2f a = { prow[kb], prow[kb + 1] };
#pragma unroll
        for (int nt = 0; nt < 4; ++nt) {
            v2f b = { W1[kb * HIDDEN + nt * 16 + sub],
                      W1[(kb + 1) * HIDDEN + nt * 16 + sub] };
            acc[nt] = __builtin_amdgcn_wmma_f32_16x16x4_f32(
                false, a, false, b, (short)0, acc[nt], false, false);
        }
    }
#pragma unroll
    for (int nt = 0; nt < 4; ++nt)
#pragma unroll
        for (int i = 0; i < 8; ++i)
            X[(i + half * 8) * LDS_STRIDE + nt * 16 + sub] = silu(acc[nt][i]);

    // ---------------- Layer 2: [16x64] @ [64x64] + b2, SiLU ----------------
#pragma unroll
    for (int nt = 0; nt < 4; ++nt) {
        const float bv = b2[nt * 16 + sub];
#pragma unroll
        for (int i = 0; i < 8; ++i) acc[nt][i] = bv;
    }
#pragma unroll 4
    for (int ks = 0; ks < 16; ++ks) {
        const int kb = ks * 4 + half * 2;
        v2f a = *(const v2f*)&X[sub * LDS_STRIDE + kb];   // 8B-aligned ds load
#pragma unroll
        for (int nt = 0; nt < 4; ++nt) {
            v2f b = { W2[kb * HIDDEN + nt * 16 + sub],
                      W2[(kb + 1) * HIDDEN + nt * 16 + sub] };
            acc[nt] = __builtin_amdgcn_wmma_f32_16x16x4_f32(
                false, a, false, b, (short)0, acc[nt], false, false);
        }
    }
#pragma unroll
    for (int nt = 0; nt < 4; ++nt)
#pragma unroll
        for (int i = 0; i < 8; ++i)
            X[(i + half * 8) * LDS_STRIDE + nt * 16 + sub] = silu(acc[nt][i]);

    // ---------------- Layer 3: [16x64] @ [64x1] + b3 (per-row dot) ----------
    if (lane < 16) {                          // after last WMMA: predication OK
        float s = 0.f;
#pragma unroll 8
        for (int c = 0; c < HIDDEN; ++c)
            s += X[lane * LDS_STRIDE + c] * W3[c];
        const int row = tile * 16 + lane;
        if (row < B) out[row] = s + b3[0];
    }
}

extern "C" void kernel_launch(void* const* d_in, const int* in_sizes, int n_in,
                              void* d_out, int out_size, void* d_ws, size_t ws_size,
                              hipStream_t stream) {
    const float* h   = (const float*)d_in[0];
    const int*   ids = (const int*)d_in[1];
    // d_in[2] = num_segments (device scalar) — B is known on host via out_size
    const float* W1  = (const float*)d_in[3];
    const float* b1  = (const float*)d_in[4];
    const float* W2  = (const float*)d_in[5];
    const float* b2  = (const float*)d_in[6];
    const float* W3  = (const float*)d_in[7];
    const float* b3  = (const float*)d_in[8];
    float*       out = (float*)d_out;
    float*    pooled = (float*)d_ws;          // B * 128 floats (~25.6 MB)

    const int n_total = in_sizes[0] / D_MODEL;
    const int B       = out_size;

    const int wavesPerBlock = 8;              // 256 threads = 8 wave32
    const int blocks1 = (B + wavesPerBlock - 1) / wavesPerBlock;
    pool_kernel<<<blocks1, 256, 0, stream>>>(h, ids, pooled, n_total, B);

    const int tiles   = (B + 15) / 16;
    const int blocks2 = (tiles + wavesPerBlock - 1) / wavesPerBlock;
    mlp_kernel<<<blocks2, 256, 0, stream>>>(pooled, W1, b1, W2, b2, W3, b3,
                                            out, tiles, B);
}